// GraphShiftOperator_19774029431556
// MI455X (gfx1250) — compile-verified
//
#include <hip/hip_runtime.h>

#define MATN 8192
constexpr float EPS = 1e-8f;

typedef float v2f __attribute__((ext_vector_type(2)));
typedef float v8f __attribute__((ext_vector_type(8)));

#if __has_builtin(__builtin_amdgcn_wmma_f32_16x16x4_f32)
#define USE_WMMA44 1
#endif

#if __has_builtin(__builtin_amdgcn_global_load_async_to_lds_b32)
#define USE_ASYNC_LDS 1
#endif

#define GLOBAL_AS __attribute__((address_space(1)))
#define LDS_AS    __attribute__((address_space(3)))

static __device__ __forceinline__ void wait_async_zero() {
#if __has_builtin(__builtin_amdgcn_s_wait_asynccnt)
  __builtin_amdgcn_s_wait_asynccnt(0);
#else
  asm volatile("s_wait_asynccnt 0" ::: "memory");
#endif
}

// Scratch layout (floats):
//   rowpart : 8  * 8192  at offset 0        (per col-slab row partials)
//   colpart : 32 * 8192  at offset 65536    (per row-slab col partials)
//   dinv    : 2  * 8192  at offset 327680   (dinv_out | dinv_in)
#define WS_ROWPART 0
#define WS_COLPART 65536
#define WS_DINV    327680

// ---------------------------------------------------------------- column sums
// grid (32, 32), block 256: thread-per-column, 256-row slab per block.
// Deterministic: each slab writes its own partial; no atomics.
__global__ __launch_bounds__(256) void colsum_kernel(const float* __restrict__ A,
                                                     float* __restrict__ colpart) {
  int col = blockIdx.x * 256 + threadIdx.x;
  size_t row0 = (size_t)blockIdx.y * 256;
  const float* p = A + row0 * MATN + col;
  float acc = 0.0f;
#pragma unroll 8
  for (int r = 0; r < 256; ++r) acc += p[(size_t)r * MATN];
  colpart[(size_t)blockIdx.y * MATN + col] = acc;
}

// ---------------------------------------------------------------- row sums (WMMA)
// grid (8 col-slabs, 64 row-blocks), block 256 (8 waves).
// Each wave owns 16 rows; A-operand of V_WMMA_F32_16X16X4_F32 loaded per the
// documented layout: lane L -> row L%16, K pair 2*(L/16). B = ones => D rows
// accumulate exact fp32 row sums. Partials per col-slab; no atomics.
__global__ __launch_bounds__(256) void rowsum_kernel(const float* __restrict__ A,
                                                     float* __restrict__ rowpart) {
  int lane = threadIdx.x & 31;
  int wave = threadIdx.x >> 5;
  int rowbase = (blockIdx.y * 8 + wave) * 16;
  int row = rowbase + (lane & 15);
  int coff = (lane >> 4) << 1;  // lanes 16-31 carry K=2,3
  const float* base = A + (size_t)row * MATN + (size_t)blockIdx.x * 1024 + coff;

#ifdef USE_WMMA44
  v8f acc0 = {};
  v8f acc1 = {};
  v2f ones = {1.0f, 1.0f};
#pragma unroll 4
  for (int k = 0; k < 1024; k += 8) {
    v2f a0 = *(const v2f*)(base + k);
    v2f a1 = *(const v2f*)(base + k + 4);
    acc0 = __builtin_amdgcn_wmma_f32_16x16x4_f32(false, a0, false, ones,
                                                 (short)0, acc0, false, false);
    acc1 = __builtin_amdgcn_wmma_f32_16x16x4_f32(false, a1, false, ones,
                                                 (short)0, acc1, false, false);
  }
  // C/D layout: VGPR v, lanes 0-15 -> row v, lanes 16-31 -> row v+8 (col = lane%16).
  if ((lane & 15) == 0) {
    int rb = rowbase + (lane >> 4) * 8;
    float* dst = rowpart + (size_t)blockIdx.x * MATN + rb;
#pragma unroll
    for (int v = 0; v < 8; ++v) dst[v] = acc0[v] + acc1[v];
  }
#else
  float acc = 0.0f;
  for (int k = 0; k < 1024; k += 4) {
    v2f a = *(const v2f*)(base + k);
    acc += a.x + a.y;
  }
  acc += __shfl_xor(acc, 16, 32);
  if (lane < 16) rowpart[(size_t)blockIdx.x * MATN + row] = acc;
#endif
}

// ---------------------------------------------------------------- fold + reciprocal
// Deterministic fixed-order fold of partials, then 1/(deg+eps).
// grid 64, block 256: i < 8192 -> dinv_out, else dinv_in.
__global__ __launch_bounds__(256) void reduce_invert_kernel(
    const float* __restrict__ rowpart, const float* __restrict__ colpart,
    float* __restrict__ dinv) {
  int i = blockIdx.x * 256 + threadIdx.x;
  if (i < MATN) {
    float s = 0.0f;
#pragma unroll
    for (int p = 0; p < 8; ++p) s += rowpart[(size_t)p * MATN + i];
    dinv[i] = 1.0f / (s + EPS);
  } else {
    int j = i - MATN;
    float s = 0.0f;
#pragma unroll
    for (int p = 0; p < 32; ++p) s += colpart[(size_t)p * MATN + j];
    dinv[MATN + j] = 1.0f / (s + EPS);
  }
}

// ---------------------------------------------------------------- scale + transpose
// grid (128,128), block (64,4). 64x64 tile, LDS pad 65 -> conflict-free transpose.
// rev[i][j] = A[i][j]*dinv_in[j];  fwd[p][q] = A[q][p]*dinv_out[q].
// Outputs are write-once streams: non-temporal stores keep them out of L2.
__global__ __launch_bounds__(256) void scale_kernel(const float* __restrict__ A,
                                                    const float* __restrict__ dinv_out,
                                                    const float* __restrict__ dinv_in,
                                                    float* __restrict__ fwd,
                                                    float* __restrict__ rev) {
  __shared__ float t[64][65];
  int row0 = blockIdx.y * 64;
  int col0 = blockIdx.x * 64;
  int tx = threadIdx.x;  // 0..63
  int ty = threadIdx.y;  // 0..3
  float din = dinv_in[col0 + tx];
  float dout = dinv_out[row0 + tx];

#ifdef USE_ASYNC_LDS
  // Async global->LDS with per-lane LDS address: the transpose happens inside
  // the async scatter; tracked on ASYNCcnt.
#pragma unroll
  for (int rr = 0; rr < 16; ++rr) {
    int lr = ty * 16 + rr;
    const float* g = A + (size_t)(row0 + lr) * MATN + col0 + tx;
    __builtin_amdgcn_global_load_async_to_lds_b32(
        (GLOBAL_AS int*)g, (LDS_AS int*)&t[tx][lr], 0, 0);
  }
  wait_async_zero();
  __syncthreads();
#pragma unroll
  for (int rr = 0; rr < 16; ++rr) {
    int lr = ty * 16 + rr;
    // t[tx][lr] = A[row0+lr][col0+tx];  t[lr][tx] = A[row0+tx][col0+lr]
    __builtin_nontemporal_store(t[tx][lr] * din,
                                &rev[(size_t)(row0 + lr) * MATN + col0 + tx]);
    __builtin_nontemporal_store(t[lr][tx] * dout,
                                &fwd[(size_t)(col0 + lr) * MATN + row0 + tx]);
  }
#else
#pragma unroll
  for (int rr = 0; rr < 16; ++rr) {
    int lr = ty * 16 + rr;
    float v = A[(size_t)(row0 + lr) * MATN + col0 + tx];
    __builtin_nontemporal_store(v * din,
                                &rev[(size_t)(row0 + lr) * MATN + col0 + tx]);
    t[tx][lr] = v;  // transposed store, stride 65 -> conflict-free
  }
  __syncthreads();
#pragma unroll
  for (int rr = 0; rr < 16; ++rr) {
    int li = ty * 16 + rr;
    __builtin_nontemporal_store(t[li][tx] * dout,
                                &fwd[(size_t)(col0 + li) * MATN + row0 + tx]);
  }
#endif
}

// ---------------------------------------------------------------- launch
extern "C" void kernel_launch(void* const* d_in, const int* in_sizes, int n_in,
                              void* d_out, int out_size, void* d_ws, size_t ws_size,
                              hipStream_t stream) {
  const float* A = (const float*)d_in[0];
  float* out = (float*)d_out;  // [fwd (N*N) | rev (N*N)]
  float* ws = (float*)d_ws;

  float* rowpart = ws + WS_ROWPART;  // 8  * 8192
  float* colpart = ws + WS_COLPART;  // 32 * 8192
  float* dinv = ws + WS_DINV;        // dinv_out (8192) | dinv_in (8192)

  // 1) column sums (coalesced stream, per-slab partials)
  colsum_kernel<<<dim3(32, 32), 256, 0, stream>>>(A, colpart);
  // 2) row sums via f32 WMMA against ones (per-slab partials)
  rowsum_kernel<<<dim3(8, 64), 256, 0, stream>>>(A, rowpart);
  // 3) deterministic fixed-order fold + reciprocals
  reduce_invert_kernel<<<64, 256, 0, stream>>>(rowpart, colpart, dinv);
  // 4) fused scale + transpose (dominant pass: 256 MiB read, 512 MiB NT write)
  scale_kernel<<<dim3(128, 128), dim3(64, 4), 0, stream>>>(
      A, dinv, dinv + MATN, out, out + (size_t)MATN * MATN);
}